// PredictiveWorldModel_36790689857564
// MI455X (gfx1250) — compile-verified
//
#include <hip/hip_runtime.h>
#include <hip/hip_bf16.h>
#include <math.h>

typedef __attribute__((ext_vector_type(2))) float v2f;
typedef __attribute__((ext_vector_type(4))) float v4f;
typedef __attribute__((ext_vector_type(8))) float v8f;

#define D_DET   2048
#define D_OBS   4096
#define N_CAT   32
#define N_CLS   32
#define CACHE_N 8
#define D_STOCH 1024
#define D_IN    5120
#define D_LAT   3072

#define WAVES   16      // waves per block (K-split factor)
#define TPB     512     // threads per block for GEMV/GRU

// ---- output layout (flat concat, return order) ----
#define OUT_H        0
#define OUT_Z        2048
#define OUT_PRED     3072
#define OUT_PRED_ERR 7168
#define OUT_KL       7169
#define OUT_RECON    7170
#define OUT_KEYS     7171
#define OUT_VALS     (7171 + 32768)
#define OUT_VALID    (7171 + 65536)

// ---- workspace layout (float units) ----
#define WS_WIDX  0
#define WS_PH1   64
#define WS_QH1   (64 + 2048)
#define WS_PLOG  (64 + 4096)
#define WS_QLOG  (64 + 5120)
#define WS_DH1   (64 + 6144)

__device__ __forceinline__ v8f wmma4(v2f a, v2f b, v8f c) {
    // V_WMMA_F32_16X16X4_F32 : D(16x16,f32) = A(16x4,f32) * B(4x16,f32) + C
    return __builtin_amdgcn_wmma_f32_16x16x4_f32(false, a, false, b, (short)0, c,
                                                 false, false);
}

// Two chained wmmas consuming one 16B-aligned float4 of A and B per lane.
// K-slot permutation (halves hold {k..k+3} / {k+4..k+7}) is consistent
// between A and B, so the accumulated dot product is exact.
__device__ __forceinline__ v8f wmma8(v4f a, v4f b, v8f c) {
    v2f a0 = {a.x, a.y}, b0 = {b.x, b.y};
    c = wmma4(a0, b0, c);
    v2f a1 = {a.z, a.w}, b1 = {b.z, b.w};
    return wmma4(a1, b1, c);
}

__device__ __forceinline__ float elu1(float v) { return v > 0.f ? v : (expf(v) - 1.f); }
__device__ __forceinline__ float symlogf(float x) { return copysignf(log1pf(fabsf(x)), x); }

// =====================================================================
// Generic fused GEMV: out[m] = act( W[m,:] . concat(x1[0:n1], x2) + bias[m] )
// Block = 512 threads = 16 waves over ONE 16-row tile; each wave covers
// K/16 of the dot with b128 loads + v_wmma_f32_16x16x4_f32 pairs.
// B operand is broadcast across N so every lane's acc[j] holds the row dot.
// Requires: K % 128 == 0, n1 % 4 == 0, all pointers 16B aligned.
// =====================================================================
__global__ __launch_bounds__(TPB)
void gemv_wmma(const float* __restrict__ W, const float* __restrict__ x1, int n1,
               const float* __restrict__ x2, const float* __restrict__ bias,
               float* __restrict__ out, int K, int act)
{
    __shared__ float sm[WAVES][16];
    const int base = blockIdx.x * 16;
    const int wave = threadIdx.x >> 5;
    const int lane = threadIdx.x & 31;
    const int half = lane >> 4;
    const int mrow = lane & 15;
    const int Kc   = K >> 4;             // per-wave K chunk
    const int kbeg = wave * Kc;
    const int kend = kbeg + Kc;
    const float* __restrict__ wr = W + (size_t)(base + mrow) * (size_t)K;

    v8f acc = {};
    for (int k = kbeg; k < kend; k += 8) {
        const int kk = k + (half << 2);          // 16B-aligned float4 offset
        const v4f b = (kk < n1) ? *(const v4f*)(x1 + kk)
                                : *(const v4f*)(x2 + kk - n1);
        const v4f a = *(const v4f*)(wr + kk);
        acc = wmma8(a, b, acc);
    }
    // lanes 0 / 16 carry rows base+0..7 / base+8..15 in acc[0..7]
    if (mrow == 0) {
        #pragma unroll
        for (int j = 0; j < 8; ++j) sm[wave][half * 8 + j] = acc[j];
    }
    __syncthreads();
    if (threadIdx.x < 16) {
        float s = 0.f;
        #pragma unroll
        for (int w = 0; w < WAVES; ++w) s += sm[w][threadIdx.x];
        s += bias[base + threadIdx.x];
        if (act) s = elu1(s);
        out[base + threadIdx.x] = s;
    }
}

// =====================================================================
// Fused GRU cell: block owns 16 rows of h_new; computes all six dots
// (i_r,i_z,i_n over x=concat(z,obs); h_r,h_z,h_n over h) with WMMA,
// then applies the gate math. Writes h_new directly into d_out.
// =====================================================================
__global__ __launch_bounds__(TPB)
void gru_wmma(const float* __restrict__ Wih, const float* __restrict__ Whh,
              const float* __restrict__ bih, const float* __restrict__ bhh,
              const float* __restrict__ z,   const float* __restrict__ obs,
              const float* __restrict__ h,   float* __restrict__ hnew)
{
    __shared__ float sm[6][WAVES][16];
    const int base = blockIdx.x * 16;
    const int wave = threadIdx.x >> 5;
    const int lane = threadIdx.x & 31;
    const int half = lane >> 4;
    const int mrow = lane & 15;

    const float* __restrict__ wir = Wih + (size_t)(base        + mrow) * D_IN;
    const float* __restrict__ wiz = Wih + (size_t)(base + 2048 + mrow) * D_IN;
    const float* __restrict__ win = Wih + (size_t)(base + 4096 + mrow) * D_IN;

    v8f aR = {}, aZ = {}, aN = {};
    {
        const int Kc = D_IN >> 4;                   // 320
        const int kbeg = wave * Kc, kend = kbeg + Kc;
        for (int k = kbeg; k < kend; k += 8) {
            const int kk = k + (half << 2);
            const v4f b = (kk < D_STOCH) ? *(const v4f*)(z + kk)
                                         : *(const v4f*)(obs + kk - D_STOCH);
            aR = wmma8(*(const v4f*)(wir + kk), b, aR);
            aZ = wmma8(*(const v4f*)(wiz + kk), b, aZ);
            aN = wmma8(*(const v4f*)(win + kk), b, aN);
        }
    }
    const float* __restrict__ whr = Whh + (size_t)(base        + mrow) * D_DET;
    const float* __restrict__ whz = Whh + (size_t)(base + 2048 + mrow) * D_DET;
    const float* __restrict__ whn = Whh + (size_t)(base + 4096 + mrow) * D_DET;

    v8f hR = {}, hZ = {}, hN = {};
    {
        const int Kc = D_DET >> 4;                  // 128
        const int kbeg = wave * Kc, kend = kbeg + Kc;
        for (int k = kbeg; k < kend; k += 8) {
            const int kk = k + (half << 2);
            const v4f b = *(const v4f*)(h + kk);
            hR = wmma8(*(const v4f*)(whr + kk), b, hR);
            hZ = wmma8(*(const v4f*)(whz + kk), b, hZ);
            hN = wmma8(*(const v4f*)(whn + kk), b, hN);
        }
    }
    if (mrow == 0) {
        #pragma unroll
        for (int j = 0; j < 8; ++j) {
            const int r = half * 8 + j;
            sm[0][wave][r] = aR[j]; sm[1][wave][r] = aZ[j]; sm[2][wave][r] = aN[j];
            sm[3][wave][r] = hR[j]; sm[4][wave][r] = hZ[j]; sm[5][wave][r] = hN[j];
        }
    }
    __syncthreads();
    if (threadIdx.x < 16) {
        const int row = base + threadIdx.x;
        float ir = 0.f, iz = 0.f, in_ = 0.f, hr = 0.f, hz = 0.f, hn = 0.f;
        #pragma unroll
        for (int w = 0; w < WAVES; ++w) {
            ir += sm[0][w][threadIdx.x]; iz += sm[1][w][threadIdx.x];
            in_ += sm[2][w][threadIdx.x];
            hr += sm[3][w][threadIdx.x]; hz += sm[4][w][threadIdx.x];
            hn += sm[5][w][threadIdx.x];
        }
        ir += bih[row]; iz += bih[row + 2048]; in_ += bih[row + 4096];
        hr += bhh[row]; hz += bhh[row + 2048]; hn += bhh[row + 4096];
        const float r  = 1.f / (1.f + expf(-(ir + hr)));
        const float zg = 1.f / (1.f + expf(-(iz + hz)));
        const float n  = tanhf(in_ + r * hn);
        hnew[row] = (1.f - zg) * n + zg * h[row];
    }
}

// =====================================================================
// Gumbel-max straight-through sample + KL. 32 waves, one category each;
// lane = class. Wave-level shuffle reductions (wave32).
// =====================================================================
__global__ __launch_bounds__(1024)
void sample_kl(const float* __restrict__ plog, const float* __restrict__ qlog,
               const float* __restrict__ u, float* __restrict__ out)
{
    __shared__ float klsh[32];
    const int cat = threadIdx.x >> 5;
    const int lane = threadIdx.x & 31;
    const float lq = qlog[cat * 32 + lane];
    const float lp = plog[cat * 32 + lane];
    const float g  = -logf(-logf(u[cat * 32 + lane] + 1e-9f) + 1e-9f);

    // argmax(lq + g), first-index tie-break
    float bv = lq + g; int bi = lane;
    #pragma unroll
    for (int m = 16; m; m >>= 1) {
        const float ov = __shfl_xor(bv, m, 32);
        const int   oi = __shfl_xor(bi, m, 32);
        if (ov > bv || (ov == bv && oi < bi)) { bv = ov; bi = oi; }
    }
    // softmax / log-softmax of posterior
    float mq = lq;
    #pragma unroll
    for (int m = 16; m; m >>= 1) mq = fmaxf(mq, __shfl_xor(mq, m, 32));
    const float eq = expf(lq - mq);
    float sq = eq;
    #pragma unroll
    for (int m = 16; m; m >>= 1) sq += __shfl_xor(sq, m, 32);
    const float prob  = eq / sq;
    const float oneh  = (lane == bi) ? 1.f : 0.f;
    out[OUT_Z + cat * 32 + lane] = oneh + prob - prob;   // straight-through value
    const float lpost = lq - mq - logf(sq);
    // log-softmax of prior
    float mp = lp;
    #pragma unroll
    for (int m = 16; m; m >>= 1) mp = fmaxf(mp, __shfl_xor(mp, m, 32));
    float sp = expf(lp - mp);
    #pragma unroll
    for (int m = 16; m; m >>= 1) sp += __shfl_xor(sp, m, 32);
    const float lprior = lp - mp - logf(sp);

    float kt = expf(lpost) * (lpost - lprior);
    #pragma unroll
    for (int m = 16; m; m >>= 1) kt += __shfl_xor(kt, m, 32);
    if (lane == 0) klsh[cat] = kt;
    __syncthreads();
    if (threadIdx.x < 32) {
        float v = klsh[threadIdx.x];
        #pragma unroll
        for (int m = 16; m; m >>= 1) v += __shfl_xor(v, m, 32);
        if (threadIdx.x == 0) out[OUT_KL] = v / (float)N_CAT;
    }
}

// ---- block-wide sum, valid only for blockDim == 1024 ----
__device__ __forceinline__ float block_sum(float v, float* sm)
{
    #pragma unroll
    for (int m = 16; m; m >>= 1) v += __shfl_xor(v, m, 32);
    const int w = threadIdx.x >> 5, lane = threadIdx.x & 31;
    if (lane == 0) sm[w] = v;
    __syncthreads();
    if (w == 0) {
        float r = sm[lane];
        #pragma unroll
        for (int m = 16; m; m >>= 1) r += __shfl_xor(r, m, 32);
        if (lane == 0) sm[0] = r;
    }
    __syncthreads();
    const float r = sm[0];
    __syncthreads();
    return r;
}

// =====================================================================
// Losses + cosine-similarity cache decision (single block).
// =====================================================================
__global__ __launch_bounds__(1024)
void finalize(const float* __restrict__ pred, const float* __restrict__ obs,
              const float* __restrict__ keys, const unsigned char* __restrict__ valid,
              const int* __restrict__ cidx, float* __restrict__ out,
              int* __restrict__ widx_ws)
{
    __shared__ float red[32];
    float se = 0.f, rec = 0.f, pp = 0.f;
    float dk[CACHE_N] = {}, k2[CACHE_N] = {};
    for (int i = threadIdx.x; i < D_OBS; i += 1024) {
        const float p = pred[i], o = obs[i];
        const float d = p - o;                     se  += d * d;
        const float s = symlogf(p) - symlogf(o);   rec += s * s;
        pp += p * p;
        #pragma unroll
        for (int c = 0; c < CACHE_N; ++c) {
            const float kv = keys[c * D_OBS + i];
            dk[c] += kv * p; k2[c] += kv * kv;
        }
    }
    se  = block_sum(se, red);
    rec = block_sum(rec, red);
    pp  = block_sum(pp, red);
    #pragma unroll
    for (int c = 0; c < CACHE_N; ++c) {
        dk[c] = block_sum(dk[c], red);
        k2[c] = block_sum(k2[c], red);
    }
    if (threadIdx.x == 0) {
        const float pn = sqrtf(pp);
        out[OUT_PRED_ERR] = sqrtf(se);
        out[OUT_RECON]    = rec / (float)D_OBS;
        float best = -1e30f; int bi = 0;
        #pragma unroll
        for (int c = 0; c < CACHE_N; ++c) {
            const float sim = valid[c]
                ? dk[c] / ((sqrtf(k2[c]) + 1e-8f) * (pn + 1e-8f)) : -1.f;
            if (sim > best) { best = sim; bi = c; }
        }
        const int widx = (best > 0.8f) ? bi : (cidx[0] & (CACHE_N - 1));
        *widx_ws = widx;
        #pragma unroll
        for (int c = 0; c < CACHE_N; ++c)
            out[OUT_VALID + c] = (c == widx) ? 1.f : (valid[c] ? 1.f : 0.f);
    }
}

// =====================================================================
// Ring-cache write-back: copy keys/values with row widx replaced by pred.
// =====================================================================
__global__ __launch_bounds__(256)
void cache_write(const float* __restrict__ keys, const float* __restrict__ vals,
                 const float* __restrict__ pred, const int* __restrict__ widx_ws,
                 float* __restrict__ out)
{
    const int i = blockIdx.x * blockDim.x + threadIdx.x;   // 0..65535
    const int w = *widx_ws;
    if (i < CACHE_N * D_OBS) {
        const int row = i >> 12, col = i & (D_OBS - 1);
        out[OUT_KEYS + i] = (row == w) ? pred[col] : keys[i];
    } else {
        const int j = i - CACHE_N * D_OBS;
        const int row = j >> 12, col = j & (D_OBS - 1);
        out[OUT_VALS + j] = (row == w) ? pred[col] : vals[j];
    }
}

extern "C" void kernel_launch(void* const* d_in, const int* in_sizes, int n_in,
                              void* d_out, int out_size, void* d_ws, size_t ws_size,
                              hipStream_t stream)
{
    const float* obs     = (const float*)d_in[0];
    const float* h       = (const float*)d_in[1];
    const float* z       = (const float*)d_in[2];
    const float* u_noise = (const float*)d_in[3];
    const float* gWih    = (const float*)d_in[4];
    const float* gWhh    = (const float*)d_in[5];
    const float* gBih    = (const float*)d_in[6];
    const float* gBhh    = (const float*)d_in[7];
    const float* pW1     = (const float*)d_in[8];
    const float* pB1     = (const float*)d_in[9];
    const float* pW2     = (const float*)d_in[10];
    const float* pB2     = (const float*)d_in[11];
    const float* qW1     = (const float*)d_in[12];
    const float* qB1     = (const float*)d_in[13];
    const float* qW2     = (const float*)d_in[14];
    const float* qB2     = (const float*)d_in[15];
    const float* dW1     = (const float*)d_in[16];
    const float* dB1     = (const float*)d_in[17];
    const float* dW2     = (const float*)d_in[18];
    const float* dB2     = (const float*)d_in[19];
    const float* keys    = (const float*)d_in[20];
    const float* vals    = (const float*)d_in[21];
    const unsigned char* valid = (const unsigned char*)d_in[22];
    const int*   cidx    = (const int*)d_in[23];

    float* out = (float*)d_out;
    float* ws  = (float*)d_ws;

    // 1) GRU -> h_new (d_out + OUT_H)
    gru_wmma<<<D_DET / 16, TPB, 0, stream>>>(gWih, gWhh, gBih, gBhh, z, obs, h,
                                             out + OUT_H);
    // 2) prior hidden = elu(prior_w1 @ h_new + b1)
    gemv_wmma<<<D_DET / 16, TPB, 0, stream>>>(pW1, out + OUT_H, D_DET, out + OUT_H,
                                              pB1, ws + WS_PH1, D_DET, 1);
    // 3) post hidden = elu(post_w1 @ [h_new, obs] + b1)
    gemv_wmma<<<D_DET / 16, TPB, 0, stream>>>(qW1, out + OUT_H, D_DET, obs,
                                              qB1, ws + WS_QH1, D_DET + D_OBS, 1);
    // 4) prior logits
    gemv_wmma<<<D_STOCH / 16, TPB, 0, stream>>>(pW2, ws + WS_PH1, D_DET, ws + WS_PH1,
                                                pB2, ws + WS_PLOG, D_DET, 0);
    // 5) posterior logits
    gemv_wmma<<<D_STOCH / 16, TPB, 0, stream>>>(qW2, ws + WS_QH1, D_DET, ws + WS_QH1,
                                                qB2, ws + WS_QLOG, D_DET, 0);
    // 6) Gumbel-max sample z_flat + KL
    sample_kl<<<1, 1024, 0, stream>>>(ws + WS_PLOG, ws + WS_QLOG, u_noise, out);
    // 7) decoder hidden = elu(dec_w1 @ [h_new, z_flat] + b1)
    gemv_wmma<<<D_DET / 16, TPB, 0, stream>>>(dW1, out + OUT_H, D_DET, out + OUT_Z,
                                              dB1, ws + WS_DH1, D_LAT, 1);
    // 8) predicted = dec_w2 @ dec_h1 + b2  -> d_out + OUT_PRED
    gemv_wmma<<<D_OBS / 16, TPB, 0, stream>>>(dW2, ws + WS_DH1, D_DET, ws + WS_DH1,
                                              dB2, out + OUT_PRED, D_DET, 0);
    // 9) losses + cache decision
    finalize<<<1, 1024, 0, stream>>>(out + OUT_PRED, obs, keys, valid, cidx, out,
                                     (int*)ws);
    // 10) cache write-back
    cache_write<<<(2 * CACHE_N * D_OBS) / 256, 256, 0, stream>>>(
        keys, vals, out + OUT_PRED, (int*)ws, out);
}